// lossFuncOne_84258668413462
// MI455X (gfx1250) — compile-verified
//
#include <hip/hip_runtime.h>

// Problem constants (from the reference)
#define B_    256
#define L_    200
#define Q_    2048
#define Q2_   4096                 // 2*Q, one-hot row width of `batch`
#define LM1_  199
#define ROWS_ (B_*L_)              // 51200 one-hot rows to scan
// Streaming-scan tiling: 1600 WGs x 32 rows each, TDM tiles of 4 rows (64KB),
// double-buffered => 128KB static LDS per workgroup (WGP has 320KB).
#define WGS_          1600
#define ROWS_PER_WG_  (ROWS_/WGS_)              // 32
#define TILE_ROWS_    4
#define NTILES_       (ROWS_PER_WG_/TILE_ROWS_) // 8
#define SCAN_THREADS_ (TILE_ROWS_*32)           // one wave per tile row

typedef unsigned int u32;
typedef u32 u32x4 __attribute__((ext_vector_type(4)));
typedef int  i32x8 __attribute__((ext_vector_type(8)));
typedef int  i32x4 __attribute__((ext_vector_type(4)));

// ---------------------------------------------------------------------------
// Tensor Data Mover: DMA `nrows` consecutive rows of a [*, 4096] f32 tensor
// from global memory into LDS at byte offset `lds_byte_off`.
// D# layout per CDNA5 ISA §8 (group0: count/lds_addr/global_addr/type,
// group1: data_size/dims/tile/stride). 2D tile => extended groups zero.
// Toolchain = clang-23 therock: 6-arg builtin (g0, g1, g2x4, g3x4, x8, cpol).
// ---------------------------------------------------------------------------
__device__ __forceinline__ void tdm_load_rows(u32 lds_byte_off,
                                              const float* __restrict__ gptr,
                                              int nrows) {
  unsigned long long ga = (unsigned long long)(uintptr_t)gptr;
  u32x4 g0;
  g0.x = 1u;                                         // count=1 (valid), user mode
  g0.y = lds_byte_off;                               // lds_addr[31:0]
  g0.z = (u32)(ga & 0xffffffffu);                    // global_addr[31:0]
  g0.w = (u32)((ga >> 32) & 0x1ffffffu)              // global_addr[56:32]
       | (2u << 30);                                 // type=2 ("image")

  i32x8 g1;
  g1[0] = (2 << 16);                                 // data_size=2 => 4 bytes; wg_mask=0
  g1[1] = (int)((Q2_ & 0xffff) << 16);               // tensor_dim0[15:0]=4096
  g1[2] = (int)((Q2_ >> 16) & 0xffff)                // tensor_dim0[31:16]
        | (int)(((u32)nrows & 0xffff) << 16);        // tensor_dim1[15:0]=nrows
  g1[3] = (int)((Q2_ & 0xffff) << 16);               // tensor_dim1[31:16]=0 | tile_dim0=4096
  g1[4] = (int)((u32)nrows & 0xffff);                // tile_dim1=nrows, tile_dim2=0
  g1[5] = Q2_;                                       // tensor_dim0_stride[31:0]=4096
  g1[6] = 0;                                         // stride0[47:32]=0, stride1 lo=0
  g1[7] = 0;
  i32x4 z4 = {0, 0, 0, 0};
  i32x8 z8 = {0, 0, 0, 0, 0, 0, 0, 0};
  __builtin_amdgcn_tensor_load_to_lds(g0, g1, z4, z4, z8, 0);
}

// ---------------------------------------------------------------------------
// Kernel 1: streaming one-hot scan of `batch` at HBM rate (839 MB total).
// TDM double-buffers 4-row tiles into LDS; wave w reduces row w of the tile:
//   acc = sum_q batch[row][q] * (q+1)   => col+1 if row has its one-hot, else 0
// ---------------------------------------------------------------------------
__global__ __launch_bounds__(SCAN_THREADS_) void scan_rows_kernel(
    const float* __restrict__ batch, u32* __restrict__ colp1) {
  __shared__ float lbuf[2][TILE_ROWS_ * Q2_];        // 2 x 64KB = 128KB
  const int wave = threadIdx.x >> 5;
  const int lane = threadIdx.x & 31;
  const int row0 = blockIdx.x * ROWS_PER_WG_;

  if (wave == 0) {
    tdm_load_rows((u32)(uintptr_t)&lbuf[0][0],
                  batch + (size_t)row0 * Q2_, TILE_ROWS_);
  }
  for (int i = 0; i < NTILES_; ++i) {
    const int buf = i & 1;
    if (wave == 0) {
      if (i + 1 < NTILES_) {                         // issue next tile, then wait
        tdm_load_rows((u32)(uintptr_t)&lbuf[buf ^ 1][0],
                      batch + (size_t)(row0 + (i + 1) * TILE_ROWS_) * Q2_,
                      TILE_ROWS_);
        __builtin_amdgcn_s_wait_tensorcnt(1);        // tile i landed in LDS
      } else {
        __builtin_amdgcn_s_wait_tensorcnt(0);
      }
    }
    __syncthreads();                                 // publish tile i to all waves

    const float4* rowp =
        reinterpret_cast<const float4*>(&lbuf[buf][wave * Q2_]);
    float acc = 0.f;
    float c = (float)(lane * 4 + 1);                 // weight = column index + 1
#pragma unroll 4
    for (int k = 0; k < Q2_ / 128; ++k) {            // 32 x ds_load_b128 per lane
      float4 v = rowp[k * 32 + lane];
      acc += v.x * c + v.y * (c + 1.f) + v.z * (c + 2.f) + v.w * (c + 3.f);
      c += 128.f;
    }
    // wave32 reduce (values exact: single 1.0 per row)
    for (int off = 16; off > 0; off >>= 1) acc += __shfl_xor(acc, off, 32);
    if (lane == 0)
      colp1[row0 + i * TILE_ROWS_ + wave] = (u32)(acc + 0.5f);
    __syncthreads();                                 // tile consumed -> reusable
  }
}

// ---------------------------------------------------------------------------
// Kernel 2: per-student gather + BCE.  penm[b,t] = sigmoid(logit[b,t,qid[b,t+1]])
// when (b,t+1) valid.  Writes penm*mask, a*mask, mask, and per-student mean BCE.
// ---------------------------------------------------------------------------
__global__ __launch_bounds__(256) void bce_kernel(
    const float* __restrict__ logit, const u32* __restrict__ colp1,
    float* __restrict__ out, float* __restrict__ per_student) {
  __shared__ float s_b[256];
  __shared__ float s_m[256];
  const int b = blockIdx.x;
  const int t = threadIdx.x;
  float bce = 0.f, m = 0.f;
  if (t < LM1_) {
    const u32 cp1 = colp1[b * L_ + t + 1];
    float pen = 0.f, aa = 0.f;
    if (cp1 != 0u) {
      m = 1.f;
      const u32 col = cp1 - 1u;
      const u32 qid = col & (Q_ - 1);
      const float corr = (col < (u32)Q_) ? 1.f : 0.f; // col<Q => correct one-hot
      const float x = logit[(size_t)b * L_ * Q_ + (size_t)t * Q_ + qid];
      const float p = 1.f / (1.f + expf(-x));
      const float lp  = fmaxf(logf(p), -100.f);       // torch BCELoss clamp
      const float l1p = fmaxf(log1pf(-p), -100.f);
      bce = -(corr * lp + (1.f - corr) * l1p);
      pen = p;
      aa = corr;
    }
    const size_t o = (size_t)b * LM1_ + t;
    out[1 + o] = pen;                                 // penm * mask
    out[1 + (size_t)(B_ * LM1_) + o] = aa;            // a * mask
    out[1 + 2 * (size_t)(B_ * LM1_) + o] = m;         // mask
  }
  s_b[t] = bce;
  s_m[t] = m;
  __syncthreads();
  for (int s = 128; s > 0; s >>= 1) {
    if (t < s) { s_b[t] += s_b[t + s]; s_m[t] += s_m[t + s]; }
    __syncthreads();
  }
  if (t == 0) {
    const float len = s_m[0];                         // == length[b]
    per_student[b] = (len > 0.f) ? (s_b[0] / len) : 0.f;
  }
}

// ---------------------------------------------------------------------------
// Kernel 3: deterministic final reduction of 256 per-student means -> loss.
// ---------------------------------------------------------------------------
__global__ __launch_bounds__(256) void loss_kernel(
    const float* __restrict__ per_student, float* __restrict__ out) {
  __shared__ float s[256];
  const int t = threadIdx.x;
  s[t] = per_student[t];                              // B_ == 256 exactly
  __syncthreads();
  for (int k = 128; k > 0; k >>= 1) {
    if (t < k) s[t] += s[t + k];
    __syncthreads();
  }
  if (t == 0) out[0] = s[0];
}

extern "C" void kernel_launch(void* const* d_in, const int* in_sizes, int n_in,
                              void* d_out, int out_size, void* d_ws,
                              size_t ws_size, hipStream_t stream) {
  (void)in_sizes; (void)n_in; (void)out_size; (void)ws_size;
  const float* logit = (const float*)d_in[0];   // [B, L, Q] f32
  const float* batch = (const float*)d_in[1];   // [B, L, 2Q] f32
  float* out = (float*)d_out;                   // [1 + 3*B*(L-1)] f32

  u32*   colp1       = (u32*)d_ws;                                  // 51200 u32
  float* per_student = (float*)((char*)d_ws + (size_t)ROWS_ * 4);   // 256 f32

  scan_rows_kernel<<<WGS_, SCAN_THREADS_, 0, stream>>>(batch, colp1);
  bce_kernel<<<B_, 256, 0, stream>>>(logit, colp1, out, per_student);
  loss_kernel<<<1, 256, 0, stream>>>(per_student, out);
}